// GlobalFilter_2104533975530
// MI455X (gfx1250) — compile-verified
//
#include <hip/hip_runtime.h>

typedef __attribute__((ext_vector_type(2))) float v2f;
typedef __attribute__((ext_vector_type(8))) float v8f;

#define BATCH   512
#define NSP     196      // spatial size 14x14
#define CCH     384      // channels
#define NFREQ   112      // 14 x 8 complex bins
#define MROWS   224      // 2*NFREQ (Re/Im interleaved rows)
#define KF      196      // forward K (spatial)
#define CBLK    16       // channels per block
#define NWAVES  14
#define THREADS (NWAVES * 32)

// -------- Kernel 1: build real-valued forward/inverse DFT operator matrices --------
// F  [224 x 196]: row 2f = (1/14)cos(theta), row 2f+1 = -(1/14)sin(theta)
// iF [224 x 224]: row = spatial s (rows >=196 zero-padded),
//                 col 2f = beta/14 cos, col 2f+1 = -beta/14 sin; beta=2 except w=0,7
__global__ void build_dft(float* __restrict__ F, float* __restrict__ iF) {
    int idx = blockIdx.x * blockDim.x + threadIdx.x;
    if (idx >= MROWS * MROWS) return;
    int r  = idx / MROWS;   // 0..223
    int c2 = idx % MROWS;   // 0..223
    const float scale    = 1.0f / 14.0f;
    const float twopi_14 = 0.448798950512827605f;  // 2*pi/14

    // inverse operator entry iF[r][c2]
    {
        float v = 0.0f;
        if (r < KF) {
            int p = r / 14, q = r % 14;
            int f = c2 >> 1, ri = c2 & 1;
            int h = f >> 3, w = f & 7;
            int ph = (h * p + w * q) % 14;
            float th = twopi_14 * (float)ph;
            float beta = (w == 0 || w == 7) ? 1.0f : 2.0f;
            v = beta * scale * (ri ? -__sinf(th) : __cosf(th));
        }
        iF[r * MROWS + c2] = v;
    }
    // forward operator entry F[r][c2]
    if (c2 < KF) {
        int f = r >> 1, ri = r & 1;
        int h = f >> 3, w = f & 7;
        int p = c2 / 14, q = c2 % 14;
        int ph = (h * p + w * q) % 14;
        float th = twopi_14 * (float)ph;
        F[r * KF + c2] = scale * (ri ? -__sinf(th) : __cosf(th));
    }
}

// -------- Kernel 2: fused spectral filter per (batch, 16-channel) tile --------
__global__ __launch_bounds__(THREADS) void global_filter_kernel(
    const float* __restrict__ x,    // [B][196][384]
    const float* __restrict__ cw,   // [14][8][384][2]
    const float* __restrict__ F,    // [224][196]
    const float* __restrict__ iF,   // [224][224]
    float* __restrict__ out)        // [B][196][384]
{
    __shared__ float Xs[KF * CBLK];      // 196 x 16 input tile
    __shared__ float Zs[MROWS * CBLK];   // 224 x 16 filtered spectrum

    const int b    = blockIdx.x / (CCH / CBLK);
    const int c0   = (blockIdx.x % (CCH / CBLK)) * CBLK;
    const int tid  = threadIdx.x;
    const int wave = tid >> 5;
    const int lane = tid & 31;
    const int ln   = lane & 15;
    const int hi   = lane >> 4;          // 0: K=k0,k0+1 ; 1: K=k0+2,k0+3

    // stage X tile (16 consecutive channels of one batch) into LDS, coalesced 64B rows
    for (int i = tid; i < KF * CBLK; i += THREADS) {
        int k = i >> 4, n = i & 15;
        Xs[i] = x[(size_t)(b * NSP + k) * CCH + c0 + n];
    }
    __syncthreads();

    // ---- GEMM1: Y[16w..16w+15][0..15] = F * Xs, K = 196, fp32 WMMA 16x16x4
    v8f acc = {0.f, 0.f, 0.f, 0.f, 0.f, 0.f, 0.f, 0.f};
    {
        const float* Arow = F + (size_t)(wave * 16 + ln) * KF;
        for (int k0 = 0; k0 < KF; k0 += 4) {
            const int kA = k0 + 2 * hi;
            v2f a;  a.x  = Arow[kA];            a.y  = Arow[kA + 1];
            v2f bv; bv.x = Xs[kA * CBLK + ln];  bv.y = Xs[(kA + 1) * CBLK + ln];
            acc = __builtin_amdgcn_wmma_f32_16x16x4_f32(
                false, a, false, bv, (short)0, acc, false, false);
        }
    }

    // ---- pointwise complex multiply by per-channel weight; spill Z to LDS.
    // C-layout: VGPR v <-> tile row v + 8*hi, so (Re,Im) pairs are adjacent
    // VGPRs in the same lane -> lane-local complex FMA.
    {
        const int ch = c0 + ln;
        #pragma unroll
        for (int r = 0; r < 4; ++r) {
            const int row = wave * 16 + 2 * r + 8 * hi;  // even row = Re
            const int f   = row >> 1;                    // freq bin 0..111
            const float wre = cw[((size_t)f * CCH + ch) * 2 + 0];
            const float wim = cw[((size_t)f * CCH + ch) * 2 + 1];
            const float yre = acc[2 * r], yim = acc[2 * r + 1];
            Zs[row * CBLK + ln]       = yre * wre - yim * wim;
            Zs[(row + 1) * CBLK + ln] = yre * wim + yim * wre;
        }
    }
    __syncthreads();

    // ---- GEMM2: out[16w..16w+15][0..15] = iF * Zs, K = 224
    v8f acc2 = {0.f, 0.f, 0.f, 0.f, 0.f, 0.f, 0.f, 0.f};
    {
        const float* Arow = iF + (size_t)(wave * 16 + ln) * MROWS;
        for (int k0 = 0; k0 < MROWS; k0 += 4) {
            const int kA = k0 + 2 * hi;
            v2f a;  a.x  = Arow[kA];            a.y  = Arow[kA + 1];
            v2f bv; bv.x = Zs[kA * CBLK + ln];  bv.y = Zs[(kA + 1) * CBLK + ln];
            acc2 = __builtin_amdgcn_wmma_f32_16x16x4_f32(
                false, a, false, bv, (short)0, acc2, false, false);
        }
    }

    // ---- store (rows 196..223 of the padded output are dropped)
    #pragma unroll
    for (int v = 0; v < 8; ++v) {
        const int n_sp = wave * 16 + v + 8 * hi;
        if (n_sp < NSP)
            out[(size_t)(b * NSP + n_sp) * CCH + c0 + ln] = acc2[v];
    }
}

extern "C" void kernel_launch(void* const* d_in, const int* in_sizes, int n_in,
                              void* d_out, int out_size, void* d_ws, size_t ws_size,
                              hipStream_t stream) {
    const float* x  = (const float*)d_in[0];   // [512,196,384] f32
    const float* cw = (const float*)d_in[1];   // [14,8,384,2]  f32
    float* F  = (float*)d_ws;                  // 224*196 f32
    float* iF = F + MROWS * KF;                // 224*224 f32
    float* out = (float*)d_out;

    const int total = MROWS * MROWS;
    build_dft<<<(total + 255) / 256, 256, 0, stream>>>(F, iF);

    global_filter_kernel<<<BATCH * (CCH / CBLK), THREADS, 0, stream>>>(
        x, cw, F, iF, out);
}